// BahdanauAttention_77833397338148
// MI455X (gfx1250) — compile-verified
//
#include <hip/hip_runtime.h>
#include <math.h>

// Problem constants (match reference)
#define B  16
#define S  4096
#define H  1024
#define D2 (2 * H)

typedef float v2f __attribute__((ext_vector_type(2)));
typedef float v8f __attribute__((ext_vector_type(8)));

// Hardware tanh (V_TANH_F32, a 1-cycle-issue TRANS32 op on gfx1250) instead of
// the ~25-VALU-op libm expansion: keeps the 256MB proj_key stream HBM-bound.
#if defined(__has_builtin)
#if __has_builtin(__builtin_amdgcn_tanhf)
#define FAST_TANH(x) __builtin_amdgcn_tanhf(x)
#endif
#endif
#ifndef FAST_TANH
#define FAST_TANH(x) tanhf(x)
#endif

// ---------------------------------------------------------------------------
// K1: q[b,o] = sum_h query[b,h] * Wq[o,h]   (16x1024x1024 GEMM)
// One wave per 16-wide output-channel tile; V_WMMA_F32_16X16X4_F32, K-loop.
// ISA layouts (CDNA5 ISA):
//   A 16x4:  lane L -> M = L%16; VGPR0/1 hold K = 2*(L/16) + {0,1}
//   B 4x16:  lane L -> N = L%16; VGPR0/1 hold K = 2*(L/16) + {0,1}
//   D 16x16: lane L -> N = L%16; VGPR v holds M = v + 8*(L/16)
// ---------------------------------------------------------------------------
__global__ __launch_bounds__(32)
void qproj_wmma(const float* __restrict__ query,   // [B,H] (B==16==M)
                const float* __restrict__ Wq,      // [H,H] row o, col h
                float* __restrict__ qout)          // [B,H]
{
    const int lane  = threadIdx.x & 31;
    const int half  = lane >> 4;      // 0: lanes 0-15, 1: lanes 16-31
    const int mn    = lane & 15;      // M for A, N for B/D
    const int ntile = blockIdx.x;     // 64 tiles of 16 output channels
    const int nglob = ntile * 16 + mn;

    v8f acc = {};
    for (int k0 = 0; k0 < H; k0 += 4) {
        const int k = k0 + 2 * half;
        v2f a, b;
        a.x = query[mn * H + k];
        a.y = query[mn * H + k + 1];
        // B[k][n] = Wq^T[k][n] = Wq[n][k]
        b.x = Wq[nglob * H + k];
        b.y = Wq[nglob * H + k + 1];
        acc = __builtin_amdgcn_wmma_f32_16x16x4_f32(
                /*neg_a=*/false, a, /*neg_b=*/false, b,
                /*c_mod=*/(short)0, acc, /*reuse_a=*/false, /*reuse_b=*/false);
    }
#pragma unroll
    for (int v = 0; v < 8; ++v) {
        const int m = v + 8 * half;
        qout[m * H + ntile * 16 + mn] = acc[v];
    }
}

// ---------------------------------------------------------------------------
// K2: scores[b,s] = sum_h tanh(q[b,h] + proj_key[b,s,h]) * we[h], mask -> -inf
// One wave32 per (b,s) row; 8 float4 loads per lane; shfl_xor reduction.
// Streams 256MB of proj_key — bandwidth-bound pass #1 (~11us @ 23.3 TB/s).
// ---------------------------------------------------------------------------
__global__ __launch_bounds__(256)
void scores_kernel(const float* __restrict__ pk,     // [B,S,H]
                   const float* __restrict__ q,      // [B,H]
                   const float* __restrict__ we,     // [H]
                   const int*   __restrict__ mask,   // [B,S]
                   float* __restrict__ scores)       // [B,S] (raw, pre-softmax)
{
    const int  lane = threadIdx.x & 31;
    const int  wave = threadIdx.x >> 5;
    const long row  = (long)blockIdx.x * 8 + wave;   // 0 .. B*S-1
    const int  b    = (int)(row >> 12);              // row / S (S = 4096)

    const float* pkr = pk + row * (long)H;
    const float* qr  = q + (long)b * H;

    // gfx1250 prefetch of the next slab of proj_key (global_prefetch_b8)
    __builtin_prefetch(pkr + 8 * H, 0, 1);

    float partial = 0.0f;
#pragma unroll
    for (int it = 0; it < 8; ++it) {
        const int h = it * 128 + lane * 4;
        const float4 p  = *(const float4*)(pkr + h);
        const float4 qq = *(const float4*)(qr + h);
        const float4 w  = *(const float4*)(we + h);
        partial += FAST_TANH(p.x + qq.x) * w.x;
        partial += FAST_TANH(p.y + qq.y) * w.y;
        partial += FAST_TANH(p.z + qq.z) * w.z;
        partial += FAST_TANH(p.w + qq.w) * w.w;
    }
#pragma unroll
    for (int off = 16; off; off >>= 1)
        partial += __shfl_xor(partial, off, 32);

    if (lane == 0)
        scores[row] = mask[row] ? partial : -INFINITY;
}

// ---------------------------------------------------------------------------
// K3: in-place masked softmax over S per batch; also zero the context slab
// that K4 accumulates into with atomics.
// ---------------------------------------------------------------------------
__global__ __launch_bounds__(256)
void softmax_kernel(float* __restrict__ alphas,    // [B,S] in: scores, out: softmax
                    float* __restrict__ context)   // [B,2H] zeroed here
{
    __shared__ float red[256];
    const int b   = blockIdx.x;
    const int tid = threadIdx.x;

    for (int j = tid; j < D2; j += 256)
        context[b * D2 + j] = 0.0f;

    float* row = alphas + (long)b * S;
    float vals[16];
    float mx = -INFINITY;
#pragma unroll
    for (int j = 0; j < 16; ++j) {
        vals[j] = row[tid + j * 256];
        mx = fmaxf(mx, vals[j]);
    }
    red[tid] = mx; __syncthreads();
    for (int off = 128; off; off >>= 1) {
        if (tid < off) red[tid] = fmaxf(red[tid], red[tid + off]);
        __syncthreads();
    }
    mx = red[0]; __syncthreads();

    float sum = 0.0f;
#pragma unroll
    for (int j = 0; j < 16; ++j) {
        vals[j] = __expf(vals[j] - mx);   // exp(-inf - mx) == 0 for masked
        sum += vals[j];
    }
    red[tid] = sum; __syncthreads();
    for (int off = 128; off; off >>= 1) {
        if (tid < off) red[tid] += red[tid + off];
        __syncthreads();
    }
    const float inv = 1.0f / red[0];
#pragma unroll
    for (int j = 0; j < 16; ++j)
        row[tid + j * 256] = vals[j] * inv;
}

// ---------------------------------------------------------------------------
// K4: context[b,d] = sum_s alphas[b,s] * value[b,s,d]
// Streams 512MB of value (bandwidth-bound pass #2, ~22us @ 23.3 TB/s).
// Grid (s-chunk, d-tile, b) = (16,4,16) blocks of 128 threads, each thread
// owns a float4 of d; alphas chunk staged in LDS; partials combined with
// global f32 atomics.
// ---------------------------------------------------------------------------
#define SCHUNK 256
#define DTILE  512
__global__ __launch_bounds__(128)
void context_kernel(const float* __restrict__ value,   // [B,S,2H]
                    const float* __restrict__ alphas,  // [B,S]
                    float* __restrict__ context)       // [B,2H]
{
    __shared__ float la[SCHUNK];
    const int b   = blockIdx.z;
    const int dt  = blockIdx.y;          // 0..3
    const int sc  = blockIdx.x;          // 0..15
    const int tid = threadIdx.x;         // 0..127
    const int s0  = sc * SCHUNK;

    for (int j = tid; j < SCHUNK; j += 128)
        la[j] = alphas[(long)b * S + s0 + j];
    __syncthreads();

    const int d = dt * DTILE + tid * 4;
    const float* vp = value + ((long)b * S + s0) * D2 + d;

    float4 acc = {0.f, 0.f, 0.f, 0.f};
    for (int s = 0; s < SCHUNK; ++s) {
        const float  a = la[s];
        const float4 v = *(const float4*)(vp + (long)s * D2);
        acc.x += a * v.x;
        acc.y += a * v.y;
        acc.z += a * v.z;
        acc.w += a * v.w;
    }

    float* cp = context + (long)b * D2 + d;
    atomicAdd(cp + 0, acc.x);
    atomicAdd(cp + 1, acc.y);
    atomicAdd(cp + 2, acc.z);
    atomicAdd(cp + 3, acc.w);
}

// ---------------------------------------------------------------------------
// Launch: out = [context (B*2H floats) | alphas (B*S floats)]
// ws: q buffer (B*H floats = 64 KB)
// ---------------------------------------------------------------------------
extern "C" void kernel_launch(void* const* d_in, const int* in_sizes, int n_in,
                              void* d_out, int out_size, void* d_ws, size_t ws_size,
                              hipStream_t stream)
{
    const float* query    = (const float*)d_in[0];
    const float* proj_key = (const float*)d_in[1];
    const float* value    = (const float*)d_in[2];
    const float* Wq       = (const float*)d_in[3];
    const float* we       = (const float*)d_in[4];
    const int*   mask     = (const int*)d_in[5];

    float* context = (float*)d_out;            // B*2H
    float* alphas  = (float*)d_out + B * D2;   // B*S
    float* q_ws    = (float*)d_ws;             // B*H

    qproj_wmma   <<<64, 32, 0, stream>>>(query, Wq, q_ws);
    scores_kernel<<<(B * S) / 8, 256, 0, stream>>>(proj_key, q_ws, we, mask, alphas);
    softmax_kernel<<<B, 256, 0, stream>>>(alphas, context);
    context_kernel<<<dim3(16, 4, 16), 128, 0, stream>>>(value, alphas, context);
}